// STGCN_17343077941765
// MI455X (gfx1250) — compile-verified
//
#include <hip/hip_runtime.h>
#include <hip/hip_bf16.h>
#include <cstdint>
#include <cstddef>

// ---------------------------------------------------------------------------
// ST-GCN forward for MI455X (gfx1250, wave32, WMMA f32 16x16x4).
// All heavy GEMMs (graph conv fused with adjacency, 9-tap temporal conv,
// 1x1 strided residual conv) run through V_WMMA_F32_16X16X4_F32.
// Workgroup tile 64x64 (8 waves = 4 M-tiles x 2 N-halves, 2 accumulators/wave)
// to double weight reuse vs a 64x32 tile.
// ---------------------------------------------------------------------------

typedef float v2f __attribute__((ext_vector_type(2)));
typedef float v8f __attribute__((ext_vector_type(8)));

#define EPSF   1e-5f
#define NMB    128      // N*M batch of skeletons
#define VJ     17       // joints
#define GC_KP  20       // padded per-channel reduction rows for graph GEMM (17 -> 20)
#define TC_KP  12       // padded taps for temporal conv (9 -> 12)
#define STAT_S 64       // stats split factor

// ---------------- workspace layout (float offsets) -------------------------
static const size_t OFF_A    = 0;           // 33,554,432 : activations / residual / next act
static const size_t OFF_G    = 33554432;    // 67,108,864 : graph-GEMM out, BN+ReLU in place
static const size_t OFF_T1   = 100663296;   // 33,554,432 : temporal-conv out (pre-BN)
static const size_t OFF_W    = 134217728;   //  6,815,744 : Weff (adjacency-fused weights)
static const size_t OFF_BE   = 141033472;   //      4,096 : Beff (fused bias)
static const size_t OFF_PART = 141037568;   //     24,576 : BN partial sums (reused)
static const size_t OFF_FG   = 141062144;   //        384 : graph BN scale/shift
static const size_t OFF_FT   = 141062528;   //        384 : tcn BN scale/shift
static const size_t OFF_FR   = 141062912;   //        384 : residual BN scale/shift
static const size_t OFF_FD   = 141063296;   //        160 : data_bn scale/shift
static const size_t OFF_POOL = 141063456;   //     24,576 : pooled features
// total ~141.1M floats ~ 565 MB

// ======================= data_bn (68 channels over N,T) ====================

__global__ __launch_bounds__(256) void k_dbn_partial(const float* __restrict__ x,
                                                     float* __restrict__ part) {
  // channel ch = (m, v, c); reduce over (n, t): 64*300 = 19200 elements
  const int ch = blockIdx.x, s = blockIdx.y, S = gridDim.y;
  const int m = ch / (VJ * 2), v = (ch / 2) % VJ, c = ch & 1;
  const int NT = 64 * 300, chunk = (NT + S - 1) / S;
  int beg = s * chunk, end = beg + chunk; if (end > NT) end = NT;
  float sum = 0.f, sq = 0.f;
  for (int i = beg + (int)threadIdx.x; i < end; i += 256) {
    int n = i / 300, t = i - n * 300;
    float val = x[(((size_t)(n * 2 + m) * 300 + t) * VJ + v) * 2 + c];
    sum += val; sq += val * val;
  }
  __shared__ float s1[256], s2[256];
  s1[threadIdx.x] = sum; s2[threadIdx.x] = sq; __syncthreads();
  for (int o = 128; o > 0; o >>= 1) {
    if (threadIdx.x < (unsigned)o) { s1[threadIdx.x] += s1[threadIdx.x + o]; s2[threadIdx.x] += s2[threadIdx.x + o]; }
    __syncthreads();
  }
  if (threadIdx.x == 0) { part[(ch * S + s) * 2] = s1[0]; part[(ch * S + s) * 2 + 1] = s2[0]; }
}

__global__ void k_stats_final(const float* __restrict__ part,
                              const float* __restrict__ gamma,
                              const float* __restrict__ beta,
                              float* __restrict__ F, int C, int S, float rcnt) {
  for (int c = threadIdx.x; c < C; c += blockDim.x) {
    float sum = 0.f, sq = 0.f;
    for (int s = 0; s < S; ++s) { sum += part[(c * S + s) * 2]; sq += part[(c * S + s) * 2 + 1]; }
    float mean = sum * rcnt;
    float var  = sq * rcnt - mean * mean;
    float inv  = rsqrtf(var + EPSF);
    float scale = gamma[c] * inv;
    F[2 * c]     = scale;
    F[2 * c + 1] = beta[c] - mean * scale;
  }
}

__global__ __launch_bounds__(256) void k_dbn_apply(const float* __restrict__ x,
                                                   const float* __restrict__ F,
                                                   float* __restrict__ y) {
  // y (NM, 2, 300, 17) <- normalized, transposed x (N, M, T, V, C)
  int idx = blockIdx.x * 256 + threadIdx.x;
  const int total = NMB * 2 * 300 * VJ;
  if (idx >= total) return;
  int v  = idx % VJ;
  int t  = (idx / VJ) % 300;
  int c  = (idx / (VJ * 300)) % 2;
  int nm = idx / (VJ * 300 * 2);
  int n = nm >> 1, m = nm & 1;
  int ch = (m * VJ + v) * 2 + c;
  float val = x[(((size_t)(n * 2 + m) * 300 + t) * VJ + v) * 2 + c];
  y[idx] = val * F[2 * ch] + F[2 * ch + 1];
}

// =================== adjacency-fused weight precompute =====================
// Weff[m=(c*17+w)][ci*GC_KP+k] = sum_kk gw[kk*cout+c, ci] * A[kk, k, w]  (k<17)
// Beff[m] = sum_kk gb[kk*cout+c] * sum_v A[kk, v, w]

__global__ __launch_bounds__(256) void k_weff(const float* __restrict__ gw,
                                              const float* __restrict__ gb,
                                              const float* __restrict__ A,
                                              float* __restrict__ Weff,
                                              float* __restrict__ Beff,
                                              int cin, int cout, int Mpad) {
  const int Kdim = cin * GC_KP, Mdim = cout * VJ;
  const size_t tot = (size_t)Mpad * Kdim;
  const size_t stride = (size_t)gridDim.x * 256;
  for (size_t i = blockIdx.x * 256 + threadIdx.x; i < tot; i += stride) {
    int m  = (int)(i / Kdim);
    int kc = (int)(i - (size_t)m * Kdim);
    float val = 0.f;
    if (m < Mdim) {
      int c = m / VJ, w = m - c * VJ;
      int ci = kc / GC_KP, k = kc - ci * GC_KP;
      if (k < VJ) {
        #pragma unroll
        for (int kk = 0; kk < 3; ++kk)
          val += gw[(size_t)(kk * cout + c) * cin + ci] * A[(kk * VJ + k) * VJ + w];
      }
    }
    Weff[i] = val;
  }
  for (int m = blockIdx.x * 256 + threadIdx.x; m < Mpad; m += (int)stride) {
    float b = 0.f;
    if (m < Mdim) {
      int c = m / VJ, w = m - c * VJ;
      #pragma unroll
      for (int kk = 0; kk < 3; ++kk) {
        float sA = 0.f;
        for (int v = 0; v < VJ; ++v) sA += A[(kk * VJ + v) * VJ + w];
        b += gb[kk * cout + c] * sA;
      }
    }
    Beff[m] = b;
  }
}

// ============== graph conv GEMM (WMMA f32 16x16x4), adjacency fused ========
// G[nm, c, t, w] = sum_{ci,v} Y[nm, ci, t, v] * Weff[(c,w)][(ci,v)] + Beff
// Workgroup: 8 waves, tile 64(M) x 64(N); each wave 16x32 via 2 accumulators.

__global__ __launch_bounds__(256) void k_gc_gemm(const float* __restrict__ Y,
                                                 const float* __restrict__ Weff,
                                                 const float* __restrict__ Beff,
                                                 float* __restrict__ G,
                                                 int cin, int cout, int T) {
  __shared__ float sY[64 * GC_KP];   // [t_local][k]
  __shared__ float sW[64 * GC_KP];   // [m_local][k]
  const int Mdim = cout * VJ;
  const int Kdim = cin * GC_KP;
  const int m0 = blockIdx.x * 64;
  const int t0 = blockIdx.y * 64;
  const int nm = blockIdx.z;
  const float* y = Y + (size_t)nm * cin * T * VJ;

  const int lane  = threadIdx.x & 31;
  const int wave  = threadIdx.x >> 5;
  const int mw    = wave >> 1;           // 0..3 (M sub-tile)
  const int nw    = wave & 1;            // 0..1 (32-col half)
  const int l15   = lane & 15;
  const int khalf = (lane >> 4) << 1;    // 0 or 2

  v8f acc0 = {}, acc1 = {};
  for (int c = 0; c < cin; ++c) {
    __syncthreads();
    for (int idx = threadIdx.x; idx < 64 * GC_KP; idx += 256) {
      int tl = idx / GC_KP, k = idx - tl * GC_KP;
      int t = t0 + tl;
      float val = 0.f;
      if (k < VJ && t < T) val = y[((size_t)c * T + t) * VJ + k];
      sY[idx] = val;
    }
    for (int idx = threadIdx.x; idx < 64 * GC_KP; idx += 256) {
      int ml = idx / GC_KP, k = idx - ml * GC_KP;
      sW[idx] = Weff[(size_t)(m0 + ml) * Kdim + c * GC_KP + k];
    }
    if (threadIdx.x == 0 && c + 1 < cin)
      __builtin_prefetch(&y[((size_t)(c + 1) * T + t0) * VJ], 0, 1);  // global_prefetch_b8
    __syncthreads();
    #pragma unroll
    for (int j = 0; j < GC_KP / 4; ++j) {
      int kk = 4 * j + khalf;
      v2f a  = *(const v2f*)&sW[(mw * 16 + l15) * GC_KP + kk];
      v2f b0 = *(const v2f*)&sY[(nw * 32 + l15) * GC_KP + kk];
      v2f b1 = *(const v2f*)&sY[(nw * 32 + 16 + l15) * GC_KP + kk];
      acc0 = __builtin_amdgcn_wmma_f32_16x16x4_f32(false, a, false, b0, (short)0, acc0, false, false);
      acc1 = __builtin_amdgcn_wmma_f32_16x16x4_f32(false, a, false, b1, (short)0, acc1, false, false);
    }
  }
  // epilogue: two 16-column groups per wave
  #pragma unroll
  for (int h = 0; h < 2; ++h) {
    const int tcol = t0 + nw * 32 + h * 16 + l15;
    if (tcol < T) {
      v8f acc = h ? acc1 : acc0;
      #pragma unroll
      for (int r = 0; r < 8; ++r) {
        int m = m0 + mw * 16 + r + ((lane >> 4) << 3);
        if (m < Mdim) {
          int cc = m / VJ, w = m - cc * VJ;
          G[(((size_t)nm * cout + cc) * T + tcol) * VJ + w] = acc[r] + Beff[m];
        }
      }
    }
  }
}

// ======= implicit-GEMM temporal / residual conv (WMMA f32 16x16x4) =========
// Out[nm, o, to, v] = sum_{c, dt} X[nm, c, to*stride+dt-pad, v] * W[o, c, dt] + Bias[o]
// TAPS/PAD/DTP compile-time so the WMMA chunk loop is straight-line.

template <int TAPS, int PAD, int DTP>
__global__ __launch_bounds__(256) void k_tconv_gemm(const float* __restrict__ X,
                                                    const float* __restrict__ W,
                                                    const float* __restrict__ Bias,
                                                    float* __restrict__ Out,
                                                    int cin2, int cout2, int Tin, int To,
                                                    int stride) {
  __shared__ float sX[64 * DTP];
  __shared__ float sW[64 * DTP];
  const int Ncols = To * VJ;
  const int m0 = blockIdx.x * 64;
  const int j0 = blockIdx.y * 64;
  const int nm = blockIdx.z;
  const float* x = X + (size_t)nm * cin2 * Tin * VJ;

  const int lane  = threadIdx.x & 31;
  const int wave  = threadIdx.x >> 5;
  const int mw    = wave >> 1;
  const int nw    = wave & 1;
  const int l15   = lane & 15;
  const int khalf = (lane >> 4) << 1;

  v8f acc0 = {}, acc1 = {};
  for (int c = 0; c < cin2; ++c) {
    __syncthreads();
    for (int idx = threadIdx.x; idx < 64 * DTP; idx += 256) {
      int col = idx / DTP, dt = idx - col * DTP;
      int j = j0 + col;
      float val = 0.f;
      if (dt < TAPS && j < Ncols) {
        int to = j / VJ, v = j - to * VJ;
        int ti = to * stride + dt - PAD;
        if (ti >= 0 && ti < Tin) val = x[((size_t)c * Tin + ti) * VJ + v];
      }
      sX[col * DTP + dt] = val;
    }
    for (int idx = threadIdx.x; idx < 64 * DTP; idx += 256) {
      int ml = idx / DTP, dt = idx - ml * DTP;
      int m = m0 + ml;
      float val = 0.f;
      if (dt < TAPS && m < cout2) val = W[((size_t)m * cin2 + c) * TAPS + dt];
      sW[ml * DTP + dt] = val;
    }
    __syncthreads();
    #pragma unroll
    for (int j2 = 0; j2 < (DTP >> 2); ++j2) {
      int kk = 4 * j2 + khalf;
      v2f a  = *(const v2f*)&sW[(mw * 16 + l15) * DTP + kk];
      v2f b0 = *(const v2f*)&sX[(nw * 32 + l15) * DTP + kk];
      v2f b1 = *(const v2f*)&sX[(nw * 32 + 16 + l15) * DTP + kk];
      acc0 = __builtin_amdgcn_wmma_f32_16x16x4_f32(false, a, false, b0, (short)0, acc0, false, false);
      acc1 = __builtin_amdgcn_wmma_f32_16x16x4_f32(false, a, false, b1, (short)0, acc1, false, false);
    }
  }
  #pragma unroll
  for (int h = 0; h < 2; ++h) {
    const int j = j0 + nw * 32 + h * 16 + l15;
    if (j < Ncols) {
      int to = j / VJ, v = j - to * VJ;
      v8f acc = h ? acc1 : acc0;
      #pragma unroll
      for (int r = 0; r < 8; ++r) {
        int m = m0 + mw * 16 + r + ((lane >> 4) << 3);
        if (m < cout2)
          Out[(((size_t)nm * cout2 + m) * To + to) * VJ + v] = acc[r] + Bias[m];
      }
    }
  }
}

// ===================== per-channel BN stats / apply ========================

__global__ __launch_bounds__(256) void k_stats_partial(const float* __restrict__ X,
                                                       int C, int P,
                                                       float* __restrict__ part) {
  const int c = blockIdx.x, s = blockIdx.y, S = gridDim.y;
  const int total = NMB * P;
  const int chunk = (total + S - 1) / S;
  int beg = s * chunk, end = beg + chunk; if (end > total) end = total;
  float sum = 0.f, sq = 0.f;
  for (int i = beg + (int)threadIdx.x; i < end; i += 256) {
    int nm = i / P, p = i - nm * P;
    float v = X[((size_t)nm * C + c) * P + p];
    sum += v; sq += v * v;
  }
  __shared__ float s1[256], s2[256];
  s1[threadIdx.x] = sum; s2[threadIdx.x] = sq; __syncthreads();
  for (int o = 128; o > 0; o >>= 1) {
    if (threadIdx.x < (unsigned)o) { s1[threadIdx.x] += s1[threadIdx.x + o]; s2[threadIdx.x] += s2[threadIdx.x + o]; }
    __syncthreads();
  }
  if (threadIdx.x == 0) { part[(c * S + s) * 2] = s1[0]; part[(c * S + s) * 2 + 1] = s2[0]; }
}

__global__ __launch_bounds__(256) void k_bn_relu_inplace(float* __restrict__ X,
                                                         const float* __restrict__ F,
                                                         int C, int P, int total) {
  const int stride = gridDim.x * 256;
  for (int i = blockIdx.x * 256 + threadIdx.x; i < total; i += stride) {
    int c = (i / P) % C;
    float v = X[i] * F[2 * c] + F[2 * c + 1];
    X[i] = fmaxf(v, 0.f);
  }
}

// Ynext = relu(bn(T1) + residual).  mode: 0=none, 1=identity (RES=g), 2=conv (RES=raw r)
__global__ __launch_bounds__(256) void k_final_combine(const float* __restrict__ T1,
                                                       const float* __restrict__ FT,
                                                       const float* RES,
                                                       const float* __restrict__ FR,
                                                       float* Y,
                                                       int mode, int C, int P, int total) {
  const int stride = gridDim.x * 256;
  for (int i = blockIdx.x * 256 + threadIdx.x; i < total; i += stride) {
    int c = (i / P) % C;
    float v = T1[i] * FT[2 * c] + FT[2 * c + 1];
    if (mode == 1)      v += RES[i];
    else if (mode == 2) v += RES[i] * FR[2 * c] + FR[2 * c + 1];
    Y[i] = fmaxf(v, 0.f);
  }
}

// ========================= head: pool, max, fc =============================

__global__ __launch_bounds__(128) void k_pool(const float* __restrict__ Y,
                                              float* __restrict__ pooled,
                                              int C, int plane) {
  const int nmc = blockIdx.x;
  const int nm = nmc / C, c = nmc - nm * C;
  const float* base = Y + ((size_t)nm * C + c) * plane;
  float sum = 0.f;
  for (int i = threadIdx.x; i < plane; i += 128) sum += base[i];
  __shared__ float s1[128];
  s1[threadIdx.x] = sum; __syncthreads();
  for (int o = 64; o > 0; o >>= 1) {
    if (threadIdx.x < (unsigned)o) s1[threadIdx.x] += s1[threadIdx.x + o];
    __syncthreads();
  }
  if (threadIdx.x == 0) pooled[nmc] = s1[0] / (float)plane;
}

__global__ __launch_bounds__(256) void k_head(const float* __restrict__ pooled,
                                              const float* __restrict__ fcw,
                                              const float* __restrict__ fcb,
                                              float* __restrict__ out) {
  const int n = blockIdx.x;
  float a0 = 0.f, a1 = 0.f;
  for (int c = threadIdx.x; c < 192; c += 256) {
    float h = fmaxf(pooled[(2 * n) * 192 + c], pooled[(2 * n + 1) * 192 + c]);
    a0 += h * fcw[c];
    a1 += h * fcw[192 + c];
  }
  __shared__ float s0[256], s1[256];
  s0[threadIdx.x] = a0; s1[threadIdx.x] = a1; __syncthreads();
  for (int o = 128; o > 0; o >>= 1) {
    if (threadIdx.x < (unsigned)o) { s0[threadIdx.x] += s0[threadIdx.x + o]; s1[threadIdx.x] += s1[threadIdx.x + o]; }
    __syncthreads();
  }
  if (threadIdx.x == 0) { out[2 * n] = s0[0] + fcb[0]; out[2 * n + 1] = s1[0] + fcb[1]; }
}

// ============================== host side ==================================

struct BlkPtr {
  const float *gw, *gb, *A, *gbn_g, *gbn_b, *tw, *tb, *tbn_g, *tbn_b;
  const float *rw, *rb, *rbn_g, *rbn_b;
};
struct Ptrs {
  const float *x, *dbn_g, *dbn_b, *fc_w, *fc_b;
  BlkPtr blk[6];
};

static void resolve_inputs(void* const* d_in, const int* in_sizes, Ptrs& P) {
  const bool conv[6] = {false, false, false, true, false, true};
  auto f = [&](int i) { return (const float*)d_in[i]; };
  if (in_sizes[0] == 1305600) {
    // insertion-order flatten: x, dbn_g, dbn_b, blocks..., fc_w, fc_b
    int i = 0;
    P.x = f(i++); P.dbn_g = f(i++); P.dbn_b = f(i++);
    for (int b = 0; b < 6; ++b) {
      BlkPtr& k = P.blk[b];
      k.gw = f(i++); k.gb = f(i++); k.A = f(i++); k.gbn_g = f(i++); k.gbn_b = f(i++);
      k.tw = f(i++); k.tb = f(i++); k.tbn_g = f(i++); k.tbn_b = f(i++);
      if (conv[b]) { k.rw = f(i++); k.rb = f(i++); k.rbn_g = f(i++); k.rbn_b = f(i++); }
      else         { k.rw = k.rb = k.rbn_g = k.rbn_b = nullptr; }
    }
    P.fc_w = f(i++); P.fc_b = f(i++);
  } else {
    // jax pytree (sorted-key) flatten: blocks..., dbn_b, dbn_g, fc_b, fc_w, x
    int i = 0;
    for (int b = 0; b < 6; ++b) {
      BlkPtr& k = P.blk[b];
      k.A = f(i++); k.gb = f(i++); k.gbn_b = f(i++); k.gbn_g = f(i++); k.gw = f(i++);
      if (conv[b]) { k.rb = f(i++); k.rbn_b = f(i++); k.rbn_g = f(i++); k.rw = f(i++); }
      else         { k.rw = k.rb = k.rbn_g = k.rbn_b = nullptr; }
      k.tb = f(i++); k.tbn_b = f(i++); k.tbn_g = f(i++); k.tw = f(i++);
    }
    P.dbn_b = f(i++); P.dbn_g = f(i++); P.fc_b = f(i++); P.fc_w = f(i++); P.x = f(i++);
  }
}

extern "C" void kernel_launch(void* const* d_in, const int* in_sizes, int n_in,
                              void* d_out, int out_size, void* d_ws, size_t ws_size,
                              hipStream_t stream) {
  (void)n_in; (void)out_size; (void)ws_size;
  Ptrs P;
  resolve_inputs(d_in, in_sizes, P);

  float* ws    = (float*)d_ws;
  float* bufA  = ws + OFF_A;    // activations / residual / next activations
  float* G     = ws + OFF_G;    // graph-GEMM output; BN+ReLU in place -> g
  float* T1    = ws + OFF_T1;   // temporal conv output (pre-BN)
  float* Weff  = ws + OFF_W;
  float* Beff  = ws + OFF_BE;
  float* PART  = ws + OFF_PART;
  float* FG    = ws + OFF_FG;
  float* FT    = ws + OFF_FT;
  float* FR    = ws + OFF_FR;
  float* FD    = ws + OFF_FD;
  float* POOL  = ws + OFF_POOL;
  float* out   = (float*)d_out;

  // ---- data_bn + layout transform -> bufA (NM, 2, 300, 17) ----
  k_dbn_partial<<<dim3(68, 32), 256, 0, stream>>>(P.x, PART);
  k_stats_final<<<1, 128, 0, stream>>>(PART, P.dbn_g, P.dbn_b, FD, 68, 32, 1.f / 19200.f);
  {
    int total = NMB * 2 * 300 * VJ;
    k_dbn_apply<<<(total + 255) / 256, 256, 0, stream>>>(P.x, FD, bufA);
  }

  static const int CIN[6]  = {2, 48, 48, 48, 96, 96};
  static const int COUT[6] = {48, 48, 48, 96, 96, 192};
  static const int STR[6]  = {1, 1, 1, 2, 1, 2};
  static const int RES[6]  = {0, 1, 1, 2, 1, 2};   // 0 none, 1 id, 2 conv

  int T = 300;
  for (int b = 0; b < 6; ++b) {
    const BlkPtr& K = P.blk[b];
    const int cin = CIN[b], cout = COUT[b], stride = STR[b], mode = RES[b];
    const int Mdim = cout * VJ;
    const int Mpad = ((Mdim + 63) / 64) * 64;
    const int To = (stride == 2) ? (T / 2) : T;

    // adjacency-fused weights
    k_weff<<<4096, 256, 0, stream>>>(K.gw, K.gb, K.A, Weff, Beff, cin, cout, Mpad);

    // graph conv GEMM -> G (NM, cout, T, 17)
    k_gc_gemm<<<dim3(Mpad / 64, (T + 63) / 64, NMB), 256, 0, stream>>>(
        bufA, Weff, Beff, G, cin, cout, T);

    // BN + ReLU on G (in place)
    {
      int Pplane = T * VJ, total = NMB * cout * Pplane;
      k_stats_partial<<<dim3(cout, STAT_S), 256, 0, stream>>>(G, cout, Pplane, PART);
      k_stats_final<<<1, 256, 0, stream>>>(PART, K.gbn_g, K.gbn_b, FG, cout, STAT_S,
                                           1.f / (float)(NMB * Pplane));
      k_bn_relu_inplace<<<16384, 256, 0, stream>>>(G, FG, cout, Pplane, total);
    }

    // temporal conv (9 taps, pad 4) -> T1 (NM, cout, To, 17)
    k_tconv_gemm<9, 4, TC_KP><<<dim3((cout + 63) / 64, (To * VJ + 63) / 64, NMB), 256, 0, stream>>>(
        G, K.tw, K.tb, T1, cout, cout, T, To, stride);
    {
      int Pplane = To * VJ;
      k_stats_partial<<<dim3(cout, STAT_S), 256, 0, stream>>>(T1, cout, Pplane, PART);
      k_stats_final<<<1, 256, 0, stream>>>(PART, K.tbn_g, K.tbn_b, FT, cout, STAT_S,
                                           1.f / (float)(NMB * Pplane));
    }

    // residual
    const float* resPtr = nullptr;
    if (mode == 2) {
      // 1x1 strided conv of g -> bufA (input activations there are dead now)
      k_tconv_gemm<1, 0, 4><<<dim3((cout + 63) / 64, (To * VJ + 63) / 64, NMB), 256, 0, stream>>>(
          G, K.rw, K.rb, bufA, cout, cout, T, To, stride);
      int Pplane = To * VJ;
      k_stats_partial<<<dim3(cout, STAT_S), 256, 0, stream>>>(bufA, cout, Pplane, PART);
      k_stats_final<<<1, 256, 0, stream>>>(PART, K.rbn_g, K.rbn_b, FR, cout, STAT_S,
                                           1.f / (float)(NMB * Pplane));
      resPtr = bufA;        // elementwise read-then-write aliasing with output: safe
    } else if (mode == 1) {
      resPtr = G;           // identity residual = g (stride 1, To == T)
    }

    // Ynext = relu(bn(T1) + res) -> bufA
    {
      int Pplane = To * VJ, total = NMB * cout * Pplane;
      k_final_combine<<<16384, 256, 0, stream>>>(T1, FT, resPtr, FR, bufA,
                                                 mode, cout, Pplane, total);
    }
    T = To;
  }

  // ---- head: global average pool (T=75, V=17), max over persons, fc ----
  k_pool<<<NMB * 192, 128, 0, stream>>>(bufA, POOL, 192, 75 * VJ);
  k_head<<<64, 256, 0, stream>>>(POOL, P.fc_w, P.fc_b, out);
}